// Wav2Vec2Quantizer_37314675867717
// MI455X (gfx1250) — compile-verified
//
#include <hip/hip_runtime.h>
#include <hip/hip_bf16.h>

// ---- problem constants (from reference) ----
#define GQ     2        // groups
#define NPOS   32000    // B*S = 16*2000
#define KCODES 320      // codes per group
#define DDIM   128      // dims per group
#define HDIM   256      // total hidden dim
#define KT     20       // KCODES / 16
#define CV_PAD 132      // padded LDS row stride (floats); 132 mod 64 = 4 -> conflict-free b64 reads
#define STRIPS_PER_BLOCK 8

typedef float v2f __attribute__((ext_vector_type(2)));
typedef float v4f __attribute__((ext_vector_type(4)));
typedef float v8f __attribute__((ext_vector_type(8)));

// ------------------------------------------------------------------
// Kernel 1: distances[g][n][k] = |h|^2 - 2*h.c + |c|^2 via f32 WMMA.
// One block = 8 waves; each wave computes a 16-row x 320-col strip.
// Codebook for the block's group staged in LDS (~170KB of the 320KB WGP LDS).
// Per k-step: batch all 20 B-fragment LDS loads, then 20 back-to-back WMMAs,
// so the wave is WMMA-issue-bound instead of ds-latency-bound.
// ------------------------------------------------------------------
__global__ __launch_bounds__(256) void vq_dist_kernel(
    const float* __restrict__ hs,   // [NPOS, HDIM]
    const float* __restrict__ cv,   // [GQ, KCODES, DDIM]
    float* __restrict__ dist)       // [GQ, NPOS, KCODES]
{
  extern __shared__ float smem[];
  float* scv = smem;                       // [KCODES][CV_PAD]
  float* sc2 = smem + KCODES * CV_PAD;     // [KCODES]
  float* sx2 = sc2 + KCODES;               // [8][16] per-wave row norms

  const int tid  = threadIdx.x;
  const int wave = tid >> 5;
  const int lane = tid & 31;
  const int blocks_per_g = NPOS / 16 / STRIPS_PER_BLOCK;   // 250
  const int g    = blockIdx.x / blocks_per_g;
  const int sb   = blockIdx.x % blocks_per_g;
  const int row0 = (sb * STRIPS_PER_BLOCK + wave) * 16;

  const float* cvg = cv + (size_t)g * KCODES * DDIM;

  // Stage the group's codebook into LDS (padded rows).
  for (int i = tid; i < KCODES * (DDIM / 4); i += 256) {
    const int k  = i >> 5;        // code index
    const int dc = i & 31;        // float4 chunk within row
    const v4f v = ((const v4f*)(cvg + (size_t)k * DDIM))[dc];
    float* drow = scv + k * CV_PAD + dc * 4;
    drow[0] = v.x; drow[1] = v.y; drow[2] = v.z; drow[3] = v.w;
  }
  __syncthreads();

  // Code norms c2[k] from LDS.
  for (int k = tid; k < KCODES; k += 256) {
    const float* r = scv + k * CV_PAD;
    float s = 0.f;
    #pragma unroll 8
    for (int d = 0; d < DDIM; ++d) s += r[d] * r[d];
    sc2[k] = s;
  }
  // Row norms x2 for this wave's 16 rows (lanes 0..15, one row each).
  if (lane < 16) {
    const float* hr = hs + (size_t)(row0 + lane) * HDIM + g * DDIM;
    float s = 0.f;
    for (int d = 0; d < DDIM; d += 4) {
      const v4f v = *(const v4f*)(hr + d);
      s += v.x * v.x + v.y * v.y + v.z * v.z + v.w * v.w;
    }
    sx2[wave * 16 + lane] = s;
  }
  __syncthreads();

  // 20 accumulator tiles (16x16 each) = full 320-code strip in registers.
  v8f acc[KT] = {};

  // A fragment mapping (32-bit 16x4): lane -> M = lane&15, VGPR pair -> K.
  const int m  = lane & 15;
  const int kk = (lane >> 4) * 2;
  const float* arow = hs + (size_t)(row0 + m) * HDIM + g * DDIM;

  #pragma unroll 1
  for (int kb = 0; kb < DDIM; kb += 4) {
    // Batch-load all 20 B fragments for this k-step (keeps 20 ds loads in
    // flight; compiler can use graduated s_wait_dscnt instead of wait-0
    // before every WMMA).
    v2f b[KT];
    #pragma unroll
    for (int t = 0; t < KT; ++t) {
      const float* brow = scv + (t * 16 + m) * CV_PAD + kb + kk;
      b[t].x = brow[0];
      b[t].y = brow[1];
    }
    v2f a;
    a.x = arow[kb + kk];
    a.y = arow[kb + kk + 1];
    #pragma unroll
    for (int t = 0; t < KT; ++t) {
      acc[t] = __builtin_amdgcn_wmma_f32_16x16x4_f32(
          false, a, false, b[t], (short)0, acc[t], false, false);
    }
  }

  // Epilogue: dist = x2 - 2*xc + c2; C/D layout: lane -> N, VGPR r -> M (+8 for hi lanes).
  const int n_lane = lane & 15;
  const int mbase  = (lane >> 4) * 8;
  float* dgb = dist + (size_t)g * NPOS * KCODES;
  #pragma unroll
  for (int t = 0; t < KT; ++t) {
    const float c2v = sc2[t * 16 + n_lane];
    #pragma unroll
    for (int r = 0; r < 8; ++r) {
      const int mm = mbase + r;
      const float d = sx2[wave * 16 + mm] - 2.0f * acc[t][r] + c2v;
      dgb[(size_t)(row0 + mm) * KCODES + t * 16 + n_lane] = d;
    }
  }
}

// ------------------------------------------------------------------
// Kernel 2: per (g, n): argmin over K (first-min like jnp.argmin),
// one-hot encodings, codevector gather, count accumulation.
// dist is read once -> non-temporal loads; enc/qf written once and never
// re-read -> non-temporal stores (don't evict dist from L2).
// ------------------------------------------------------------------
__global__ __launch_bounds__(256) void vq_assign_kernel(
    const float* __restrict__ cv,
    const float* __restrict__ dist,
    float* __restrict__ qf,          // [NPOS, HDIM]
    float* __restrict__ enc,         // [GQ, NPOS, KCODES]
    unsigned* __restrict__ counts)   // [GQ*KCODES]
{
  const int id = blockIdx.x * 256 + threadIdx.x;
  if (id >= GQ * NPOS) return;
  const int g = id / NPOS;
  const int n = id % NPOS;

  const float* drow = dist + ((size_t)g * NPOS + n) * KCODES;
  float best = drow[0];
  int bi = 0;
  for (int k = 0; k < KCODES; k += 4) {
    const v4f v = __builtin_nontemporal_load((const v4f*)(drow + k));
    if (v.x < best) { best = v.x; bi = k; }
    if (v.y < best) { best = v.y; bi = k + 1; }
    if (v.z < best) { best = v.z; bi = k + 2; }
    if (v.w < best) { best = v.w; bi = k + 3; }
  }

  atomicAdd(&counts[g * KCODES + bi], 1u);

  float* erow = enc + ((size_t)g * NPOS + n) * KCODES;
  const v4f z = {0.f, 0.f, 0.f, 0.f};
  for (int k = 0; k < KCODES; k += 4)
    __builtin_nontemporal_store(z, (v4f*)(erow + k));
  __builtin_nontemporal_store(1.0f, erow + bi);

  const float* crow = cv + ((size_t)g * KCODES + bi) * DDIM;
  float* q = qf + (size_t)n * HDIM + g * DDIM;
  for (int d = 0; d < DDIM; d += 4) {
    const v4f v = *(const v4f*)(crow + d);
    __builtin_nontemporal_store(v, (v4f*)(q + d));
  }
}

// ------------------------------------------------------------------
// Kernel 3: perplexity from counts.
// ------------------------------------------------------------------
__global__ void vq_perp_kernel(const unsigned* __restrict__ counts,
                               float* __restrict__ out)
{
  __shared__ float terms[GQ * KCODES];
  const int t = threadIdx.x;
  if (t < GQ * KCODES) {
    const float p = (float)counts[t] / (float)NPOS;
    terms[t] = p * logf(p + 1e-10f);
  }
  __syncthreads();
  if (t == 0) {
    float s0 = 0.f, s1 = 0.f;
    for (int k = 0; k < KCODES; ++k) { s0 += terms[k]; s1 += terms[KCODES + k]; }
    out[0] = 0.5f * (expf(-s0) + expf(-s1));
  }
}

__global__ void vq_zero_kernel(unsigned* __restrict__ counts)
{
  const int t = blockIdx.x * 256 + threadIdx.x;
  if (t < GQ * KCODES) counts[t] = 0u;
}

// ------------------------------------------------------------------
extern "C" void kernel_launch(void* const* d_in, const int* in_sizes, int n_in,
                              void* d_out, int out_size, void* d_ws, size_t ws_size,
                              hipStream_t stream) {
  const float* hs = (const float*)d_in[0];   // [16,2000,256] f32
  const float* cv = (const float*)d_in[1];   // [2,320,128]  f32

  float* out  = (float*)d_out;
  float* qf   = out;                                        // 8,192,000
  float* enc  = qf  + (size_t)NPOS * HDIM;                  // 20,480,000
  float* dist = enc + (size_t)GQ * NPOS * KCODES;           // 20,480,000
  float* perp = dist + (size_t)GQ * NPOS * KCODES;          // 1

  unsigned* counts = (unsigned*)d_ws;                       // 2560 bytes

  vq_zero_kernel<<<3, 256, 0, stream>>>(counts);

  const size_t smem = (size_t)(KCODES * CV_PAD + KCODES + 8 * 16) * sizeof(float);
  const int dist_blocks = GQ * (NPOS / 16 / STRIPS_PER_BLOCK);  // 500
  vq_dist_kernel<<<dist_blocks, 256, smem, stream>>>(hs, cv, dist);

  vq_assign_kernel<<<(GQ * NPOS + 255) / 256, 256, 0, stream>>>(cv, dist, qf, enc, counts);

  vq_perp_kernel<<<1, 640, 0, stream>>>(counts, perp);
}